// FutureEncoder_38397007626789
// MI455X (gfx1250) — compile-verified
//
#include <hip/hip_runtime.h>

typedef __attribute__((ext_vector_type(2))) float v2f;
typedef __attribute__((ext_vector_type(8))) float v8f;

#define FUTURE_K 16

// Problem dims (fixed by reference)
constexpr int Bk = 2;
constexpr int Sk = 4096;
constexpr int Hk = 1024;

constexpr int CH   = 128;        // H columns staged per LDS chunk
constexpr int NCH  = Hk / CH;    // 8 chunks
constexpr int ROWS = 33;         // rows t0 .. t0+32
constexpr int XST  = 130;        // LDS row stride (floats): even (8B align), banks spread
constexpr int GST  = 33;         // G row stride
constexpr int WST  = 34;         // W row stride (even for float2 loads)

__global__ __launch_bounds__(128) void future_encoder_kernel(
    const float* __restrict__ x, float* __restrict__ out)
{
    __shared__ float Xlds[ROWS * XST];   // staged input rows
    __shared__ float Plds[4 * 256];      // per-wave Gram partials (D layout)
    __shared__ float Glds[16 * GST];     // reduced Gram 16x32
    __shared__ float Wlds[16 * WST];     // banded softmax weights 16x32

    const int tile = blockIdx.x;              // 0 .. S/16-1
    const int b    = blockIdx.y;              // batch
    const int t0   = tile * 16;
    const int tid  = threadIdx.x;
    const int lane = tid & 31;
    const int wave = tid >> 5;
    const int lrow = lane & 15;               // M (A/D) or N (B/D) index
    const int hsel = lane >> 4;               // half-wave select

    const float* xb = x   + (size_t)b * Sk * Hk;
    float*       ob = out + (size_t)b * Sk * Hk;

    // -------- Pass 1: banded Gram G = X_tile x X_ext^T (16x32), K = 1024 --------
    v8f acc = {};                              // this wave's partial 16x16 D
    const int kbase = (wave & 1) * 64;         // K sub-range within each 128-col chunk
    const int browG = 1 + (wave >> 1) * 16;    // ext-row base: waves 0,1 -> rows 1..16; 2,3 -> 17..32

    for (int hc = 0; hc < NCH; ++hc) {
        // cooperative load rows t0..t0+32, cols [hc*CH, hc*CH+CH) into LDS (zero-fill OOB)
        for (int idx = tid; idx < ROWS * (CH / 4); idx += 128) {
            int r  = idx / (CH / 4);
            int c4 = (idx % (CH / 4)) * 4;
            float4 v = make_float4(0.f, 0.f, 0.f, 0.f);
            int rg = t0 + r;
            if (rg < Sk)
                v = *(const float4*)(xb + (size_t)rg * Hk + hc * CH + c4);
            *(float4*)&Xlds[r * XST + c4] = v;
        }
        __syncthreads();

        #pragma unroll
        for (int st = 0; st < 16; ++st) {
            int col = kbase + st * 4 + hsel * 2;
            // A: 16x4 of X_tile ; B: 4x16 of X_ext^T  (identical LDS access form)
            v2f a  = *(const v2f*)&Xlds[lrow * XST + col];
            v2f bm = *(const v2f*)&Xlds[(browG + lrow) * XST + col];
            acc = __builtin_amdgcn_wmma_f32_16x16x4_f32(
                      false, a, false, bm, (short)0, acc, false, false);
        }
        __syncthreads();
    }

    // store per-wave partial D
    #pragma unroll
    for (int i = 0; i < 8; ++i)
        Plds[wave * 256 + i * 32 + lane] = acc[i];
    __syncthreads();

    // reduce partials -> G[16][32]  (D layout: M = i + 8*(l>>4), N = l&15)
    for (int idx = tid; idx < 512; idx += 128) {
        int half = idx >> 8;       // 0: cols 0-15, 1: cols 16-31
        int rem  = idx & 255;
        int i    = rem >> 5;
        int l2   = rem & 31;
        float v = Plds[(half * 2) * 256 + rem] + Plds[(half * 2 + 1) * 256 + rem];
        int m = i + 8 * (l2 >> 4);
        int c = (l2 & 15) + 16 * half;
        Glds[m * GST + c] = v;
    }
    __syncthreads();

    // masked softmax per token -> banded weights W[16][32]
    if (tid < 16) {
        int i = tid;
        int t = t0 + i;
        for (int c = 0; c < 32; ++c) Wlds[i * WST + c] = 0.f;
        int nvalid = Sk - 1 - t;
        if (nvalid > FUTURE_K) nvalid = FUTURE_K;
        if (nvalid > 0) {
            float mx = -3.4e38f;
            for (int j = 0; j < nvalid; ++j)
                mx = fmaxf(mx, Glds[i * GST + i + j]);
            float e[FUTURE_K];
            float sum = 0.f;
            for (int j = 0; j < nvalid; ++j) {
                e[j] = __expf(Glds[i * GST + i + j] - mx);
                sum += e[j];
            }
            float inv = 1.f / sum;
            for (int j = 0; j < nvalid; ++j)
                Wlds[i * WST + i + j] = e[j] * inv;
        }
    }
    __syncthreads();

    // -------- Pass 2: out_tile = W (16x32) x X_ext (32xH) --------
    for (int hc = 0; hc < NCH; ++hc) {
        for (int idx = tid; idx < ROWS * (CH / 4); idx += 128) {
            int r  = idx / (CH / 4);
            int c4 = (idx % (CH / 4)) * 4;
            float4 v = make_float4(0.f, 0.f, 0.f, 0.f);
            int rg = t0 + r;
            if (rg < Sk)
                v = *(const float4*)(xb + (size_t)rg * Hk + hc * CH + c4);
            *(float4*)&Xlds[r * XST + c4] = v;
        }
        __syncthreads();

        #pragma unroll
        for (int nc = 0; nc < 2; ++nc) {
            int hoff = (wave * 2 + nc) * 16;   // 16-col N-chunk within this LDS chunk
            v8f d = {};
            #pragma unroll
            for (int ks = 0; ks < 8; ++ks) {
                int k0 = ks * 4 + hsel * 2;
                // A: weights 16x4
                v2f a = *(const v2f*)&Wlds[lrow * WST + k0];
                // B: X_ext 4x16  (K index k -> LDS row 1+k ; N -> column)
                v2f bm;
                bm.x = Xlds[(1 + k0) * XST + hoff + lrow];
                bm.y = Xlds[(2 + k0) * XST + hoff + lrow];
                d = __builtin_amdgcn_wmma_f32_16x16x4_f32(
                        false, a, false, bm, (short)0, d, false, false);
            }
            #pragma unroll
            for (int i = 0; i < 8; ++i) {
                ob[(size_t)(t0 + i + 8 * hsel) * Hk + hc * CH + hoff + lrow] = d[i];
            }
        }
        __syncthreads();
    }
}

extern "C" void kernel_launch(void* const* d_in, const int* in_sizes, int n_in,
                              void* d_out, int out_size, void* d_ws, size_t ws_size,
                              hipStream_t stream) {
    (void)in_sizes; (void)n_in; (void)d_ws; (void)ws_size; (void)out_size;
    const float* x = (const float*)d_in[0];
    float* out = (float*)d_out;
    dim3 grid(Sk / 16, Bk);
    dim3 block(128);
    future_encoder_kernel<<<grid, block, 0, stream>>>(x, out);
}